// Mixture_77841987272840
// MI455X (gfx1250) — compile-verified
//
#include <hip/hip_runtime.h>
#include <hip/hip_bf16.h>

#define NB       64
#define SD       191            // 3*NB - 1 spline dims
#define CUTS     32             // cuts per chunk = one wave32
#define ROWF     (CUTS * SD)    // 6112 floats per LDS buffer
#define COEF     0.936f         // 1 - MIN_BIN * NB
#define MIN_BIN  0.001f
#define MIN_D    0.001f
#define DEF_INIT 0.53974247f    // log(exp(1 - MIN_DERIVATIVE) - 1)
#define OUT_CONST (-10.5966347f) // log(0.5) - log(20000)

typedef __attribute__((address_space(3))) float lds_f;
typedef unsigned int u32x4 __attribute__((ext_vector_type(4)));
typedef unsigned int u32x8 __attribute__((ext_vector_type(8)));

__device__ __forceinline__ unsigned lds_byte_off(const float* p) {
    // generic -> LDS address space -> 32-bit LDS byte address
    return (unsigned)(size_t)(lds_f*)p;
}

__device__ __forceinline__ float softplusf(float x) {
    return fmaxf(x, 0.f) + __logf(1.f + __expf(-fabsf(x)));
}

// One TDM transfer: copy `fcount` contiguous dwords from global into LDS.
// D# (CDNA5 ISA sect. 8.3/8.4): 1-D tile, data_size = 4 bytes,
// tensor_dim0 = tile_dim0 = fcount, tile_dim1 = tensor_dim1 = 1, no padding,
// no gather, no multicast (workgroup_mask = 0), type = 2 ("image").
// Tracked with TENSORcnt; TDM ops of one wave retire in order.
__device__ __forceinline__ void tdm_load_chunk(const float* src, int fcount,
                                               unsigned lds_addr) {
    unsigned long long ga = (unsigned long long)(size_t)src;
    unsigned F = (unsigned)fcount;

    u32x4 g0;
    g0.x = 1u;                                   // count = 1 valid descriptor
    g0.y = lds_addr;                             // lds_addr [63:32]
    g0.z = (unsigned)ga;                         // global_addr [95:64]
    g0.w = (unsigned)(ga >> 32) | 0x80000000u;   // global_addr[56:32] | type=2<<30

    u32x8 g1;
    g1.s0 = 0x00020000u;                         // data_size = 2 (4 bytes)
    g1.s1 = (F & 0xffffu) << 16;                 // tensor_dim0[15:0] at bits 63:48
    g1.s2 = (F >> 16) | (1u << 16);              // tensor_dim0[31:16] | tensor_dim1=1
    g1.s3 = (F & 0xffffu) << 16;                 // tile_dim0 at bits 127:112
    g1.s4 = 1u;                                  // tile_dim1 = 1, tile_dim2 = 0
    g1.s5 = F;                                   // tensor_dim0_stride[31:0]
    g1.s6 = 0u;
    g1.s7 = 0u;

    asm volatile("tensor_load_to_lds %0, %1 th:TH_LOAD_NT"
                 :: "s"(g0), "s"(g1) : "memory");
}

__global__ void __launch_bounds__(32)
rqs_inverse_kernel(const float* __restrict__ value,
                   const float* __restrict__ delta,
                   const int*   __restrict__ genes_oi,
                   const int*   __restrict__ lgix,
                   const float* __restrict__ weight,
                   float* __restrict__ out,
                   int N, int nChunks)
{
    __shared__ float smem[2 * ROWF];
    const int lane = threadIdx.x;

    int ch = blockIdx.x;
    if (ch >= nChunks) return;

    // prime the pipeline: chunk `ch` -> buffer 0
    {
        int base = ch * CUTS;
        int nv = N - base; nv = nv < CUTS ? nv : CUTS;
        tdm_load_chunk(delta + (size_t)base * SD, nv * SD, lds_byte_off(smem));
    }

    int buf = 0;
    for (; ch < nChunks; ch += (int)gridDim.x, buf ^= 1) {
        const int chNext = ch + (int)gridDim.x;
        if (chNext < nChunks) {
            // drain our LDS stores before the TDM re-targets a buffer (WAR)
            asm volatile("s_wait_dscnt 0x0" ::: "memory");
            int nb = chNext * CUTS;
            int nv = N - nb; nv = nv < CUTS ? nv : CUTS;
            tdm_load_chunk(delta + (size_t)nb * SD, nv * SD,
                           lds_byte_off(smem + (buf ^ 1) * ROWF));
            // warm L2/WGP$ for next chunk's gathered weight row
            int c2 = nb + lane; c2 = c2 < N ? c2 : N - 1;
            const float* wr2 = weight + (size_t)genes_oi[lgix[c2]] * SD;
            __builtin_prefetch(wr2, 0, 3);
            __builtin_prefetch(wr2 + 96, 0, 3);
            // in-order retire: <=1 outstanding means this chunk's TDM is done
            asm volatile("s_wait_tensorcnt 0x1" ::: "memory");
        } else {
            asm volatile("s_wait_tensorcnt 0x0" ::: "memory");
        }

        // ---- compute this chunk (lane owns cut c) ----
        const int c = ch * CUTS + lane;
        const bool valid = (c < N);
        const int cs = valid ? c : N - 1;

        float xv = value[cs] * 1e-4f;      // ((v - A)/AB - 0.5)*2 with A=-1e4,B=1e4
        bool inside = (xv >= -1.f) && (xv <= 1.f);
        float xc = fminf(fmaxf(xv, -1.f), 1.f);

        const float* __restrict__ wr = weight + (size_t)genes_oi[lgix[cs]] * SD;
        float* lrow = smem + buf * ROWF + lane * SD;   // stride 191 -> conflict-free

        // pass 0: combine delta (LDS) + weight (L2) in-place; track softmax maxima
        float mW = -1e30f, mH = -1e30f;
#pragma unroll 8
        for (int k = 0; k < NB; ++k) {
            float s = lrow[k] + wr[k];
            lrow[k] = s;
            mW = fmaxf(mW, s);
        }
#pragma unroll 8
        for (int k = NB; k < 2 * NB; ++k) {
            float s = lrow[k] + wr[k];
            lrow[k] = s;
            mH = fmaxf(mH, s);
        }
#pragma unroll 8
        for (int k = 2 * NB; k < SD; ++k)
            lrow[k] = lrow[k] + wr[k];

        // pass 1: softmax partition sums
        float ZW = 0.f, ZH = 0.f;
#pragma unroll 8
        for (int k = 0; k < NB; ++k) ZW += __expf(lrow[k] - mW);
#pragma unroll 8
        for (int k = 0; k < NB; ++k) ZH += __expf(lrow[NB + k] - mH);
        const float invZW = COEF / ZW;
        const float invZH = COEF / ZH;

        // pass 2: streaming cumsum over heights + bin search (inverse pass bins = ch)
        float cum = 0.f, chprev = -1.f, in_ch = -1.f, in_h = 2.f;
        int idx = NB - 1;
        bool found = false;
#pragma unroll 8
        for (int k = 0; k < NB - 1; ++k) {
            float hk = MIN_BIN + __expf(lrow[NB + k] - mH) * invZH;
            cum += hk;
            float ch1 = 2.f * cum - 1.f;
            if (!found && xc < ch1) { found = true; idx = k; in_ch = chprev; in_h = ch1 - chprev; }
            chprev = ch1;
        }
        if (!found) { idx = NB - 1; in_ch = chprev; in_h = 1.f - chprev; } // last upper = 1+1e-6

        // pass 3: streaming cumsum over widths, capture bin idx
        cum = 0.f;
        float cwprev = -1.f, in_cw = -1.f, in_w = 2.f;
#pragma unroll 8
        for (int k = 0; k < NB - 1; ++k) {
            float wk = MIN_BIN + __expf(lrow[k] - mW) * invZW;
            cum += wk;
            float cw1 = 2.f * cum - 1.f;
            if (k == idx) { in_cw = cwprev; in_w = cw1 - cwprev; }
            cwprev = cw1;
        }
        if (idx == NB - 1) { in_cw = cwprev; in_w = 1.f - cwprev; }

        // derivatives at knots (ud_padded endpoints = DEFAULT_INIT)
        float ud0 = (idx == 0)      ? DEF_INIT : lrow[2 * NB + idx - 1];
        float ud1 = (idx == NB - 1) ? DEF_INIT : lrow[2 * NB + idx];
        float d0 = MIN_D + softplusf(ud0);
        float d1 = MIN_D + softplusf(ud1);

        // rational-quadratic inverse + log|det|
        float dlt  = in_h / in_w;
        float dy   = xc - in_ch;
        float s2   = d0 + d1 - 2.f * dlt;
        float a    = dy * s2 + in_h * (dlt - d0);
        float b    = in_h * d0 - dy * s2;
        float c3   = -dlt * dy;
        float disc = b * b - 4.f * a * c3;
        float root = 2.f * c3 / (-b - sqrtf(fmaxf(disc, 0.f)));
        float tom  = root * (1.f - root);
        float den  = dlt + s2 * tom;
        float omr  = 1.f - root;
        float dnum = dlt * dlt * (d1 * root * root + 2.f * dlt * tom + d0 * omr * omr);
        float logdet = 2.f * __logf(den) - __logf(dnum);

        if (valid) out[c] = OUT_CONST + (inside ? logdet : 0.f);
    }
}

extern "C" void kernel_launch(void* const* d_in, const int* in_sizes, int n_in,
                              void* d_out, int out_size, void* d_ws, size_t ws_size,
                              hipStream_t stream) {
    const float* value    = (const float*)d_in[0];
    const float* delta    = (const float*)d_in[1];
    const int*   genes_oi = (const int*)d_in[2];
    const int*   lgix     = (const int*)d_in[3];
    const float* weight   = (const float*)d_in[4];
    float* out = (float*)d_out;

    const int N = in_sizes[0];
    const int nChunks = (N + CUTS - 1) / CUTS;
    int grid = nChunks < 2048 ? nChunks : 2048;
    rqs_inverse_kernel<<<grid, 32, 0, stream>>>(value, delta, genes_oi, lgix,
                                                weight, out, N, nChunks);
}